// SSEMaskingOps_87909390614955
// MI455X (gfx1250) — compile-verified
//
#include <hip/hip_runtime.h>

// SSE partition-masking broadcast for MI455X (gfx1250, wave32).
//
// out[b,s,p,d] = x[b,s,d] * (p in partition_indices[b,s,:])
//
// Pure data movement: 16 MB read + 256 MB written  ->  ~11.7 us at 23.3 TB/s.
// The masked broadcast is a rank-1 outer product per token, computed exactly
// in f32 with V_WMMA_F32_16X16X4_F32 (A col0 = 0/1 mask, B row0 = x chunk),
// results streamed out with non-temporal b32 stores (256 MB > 192 MB L2,
// write-once -> NT keeps L2 clean).

typedef float v2f __attribute__((ext_vector_type(2)));
typedef float v8f __attribute__((ext_vector_type(8)));

#define D_MODEL   512
#define NUM_PART  16
#define TOP_K     2
#define NTILES    (D_MODEL / 16)   // 32 N-tiles of 16 columns per token
#define WAVES_PER_BLOCK 8
#define BLOCK_THREADS   (WAVES_PER_BLOCK * 32)

__global__ __launch_bounds__(BLOCK_THREADS)
void sse_mask_bcast_wmma(const float* __restrict__ x,
                         const long long* __restrict__ pidx,  // int64 indices
                         float* __restrict__ out,
                         int n_tokens)
{
    const int lane = threadIdx.x & 31;
    const int wave = threadIdx.x >> 5;
    const int t    = blockIdx.x * WAVES_PER_BLOCK + wave;   // wave-uniform
    if (t >= n_tokens) return;                              // uniform exit

    // Token's two partition indices (values in [0,16)).
    const int i0 = (int)pidx[(size_t)t * TOP_K + 0];
    const int i1 = (int)pidx[(size_t)t * TOP_K + 1];

    // A matrix (16x4 f32): lanes 0-15 VGPR0 hold A[lane][0] = mask[lane].
    // All other K columns are zero, so only a.x on the low half-wave matters.
    // Lanes >= 16 never match (indices < 16), giving 0 there automatically.
    const float m = ((lane == i0) || (lane == i1)) ? 1.0f : 0.0f;
    v2f amat;
    amat.x = m;
    amat.y = 0.0f;

    const float* __restrict__ xrow = x + (size_t)t * D_MODEL;
    float* __restrict__       orow = out + (size_t)t * (NUM_PART * D_MODEL);

    const int mrow = (lane >> 4) * 8;   // D-tile row base: 0 or 8
    const int ncol = lane & 15;         // D-tile column = lane % 16

    #pragma unroll 4
    for (int n = 0; n < NTILES; ++n) {
        // B matrix (4x16 f32): row K=0 in VGPR0 lanes 0-15 = x chunk.
        // All lanes issue the same 16-float (64 B) coalesced load; upper
        // half-wave result is masked to 0 via v_cndmask (EXEC stays all-1s).
        const float xv = xrow[n * 16 + ncol];
        v2f bmat;
        bmat.x = (lane < 16) ? xv : 0.0f;
        bmat.y = 0.0f;

        v8f acc = {};  // C = 0
        // D(16x16) = mask(16x1) (x) x_chunk(1x16), exact in f32.
        v8f d = __builtin_amdgcn_wmma_f32_16x16x4_f32(
            /*neg_a=*/false, amat, /*neg_b=*/false, bmat,
            /*c_mod=*/(short)0, acc, /*reuse_a=*/false, /*reuse_b=*/false);

        // D layout: VGPR v, lane L -> partition M = v + 8*(L>=16), col = L%16.
        float* op = orow + (size_t)mrow * D_MODEL + n * 16 + ncol;
        #pragma unroll
        for (int v = 0; v < 8; ++v) {
            __builtin_nontemporal_store(d[v], op + (size_t)v * D_MODEL);
        }
    }
}

extern "C" void kernel_launch(void* const* d_in, const int* in_sizes, int n_in,
                              void* d_out, int out_size, void* d_ws, size_t ws_size,
                              hipStream_t stream) {
    const float*     x    = (const float*)d_in[0];      // (B,S,D) f32
    const long long* pidx = (const long long*)d_in[1];  // (B,S,K) int64
    float*           out  = (float*)d_out;              // (B,S,P,D) f32

    const int n_tokens = in_sizes[0] / D_MODEL;         // B*S = 8192
    const int blocks   = (n_tokens + WAVES_PER_BLOCK - 1) / WAVES_PER_BLOCK;

    sse_mask_bcast_wmma<<<dim3(blocks), dim3(BLOCK_THREADS), 0, stream>>>(
        x, pidx, out, n_tokens);
}